// SHEncoder_53077205844617
// MI455X (gfx1250) — compile-verified
//
#include <hip/hip_runtime.h>

typedef float v2f __attribute__((ext_vector_type(2)));
typedef float v8f __attribute__((ext_vector_type(8)));

// ---------------------------------------------------------------------------
// Constant weight matrix W[20 monomials][16 SH coefficients], built so that
// out[p, j] = sum_k M[p, k] * W[k, j]   with monomial order:
//  0:1  1:x  2:y  3:z  4:xy 5:yz 6:xz 7:xx 8:yy 9:zz
// 10:x2y 11:y3 12:xyz 13:yz2 14:x2z 15:y2z 16:z3 17:xz2 18:x3 19:xy2
// ---------------------------------------------------------------------------
struct WMat { float w[20][16]; };

constexpr WMat makeW() {
    WMat m{};
    const float C0  = 0.28209479177387814f;
    const float C1  = 0.4886025119029199f;
    const float C20 = 1.0925484305920792f;
    const float C21 = -1.0925484305920792f;
    const float C22 = 0.31539156525252005f;
    const float C23 = -1.0925484305920792f;
    const float C24 = 0.5462742152960396f;
    const float C30 = -0.5900435899266435f;
    const float C31 = 2.890611442640554f;
    const float C32 = -0.4570457994644658f;
    const float C33 = 0.3731763325901154f;
    const float C34 = -0.4570457994644658f;
    const float C35 = 1.445305721320277f;
    const float C36 = -0.5900435899266435f;

    m.w[0][0]   = C0;                                        // C0 * 1
    m.w[2][1]   = -C1;                                       // -C1*y
    m.w[3][2]   = C1;                                        //  C1*z
    m.w[1][3]   = -C1;                                       // -C1*x
    m.w[4][4]   = C20;                                       // C20*xy
    m.w[5][5]   = C21;                                       // C21*yz
    m.w[9][6]   = 2.0f * C22;                                // C22*(2zz-xx-yy)
    m.w[7][6]   = -C22;
    m.w[8][6]   = -C22;
    m.w[6][7]   = C23;                                       // C23*xz
    m.w[7][8]   = C24;                                       // C24*(xx-yy)
    m.w[8][8]   = -C24;
    m.w[10][9]  = 3.0f * C30;                                // C30*y*(3xx-yy)
    m.w[11][9]  = -C30;
    m.w[12][10] = C31;                                       // C31*xyz
    m.w[13][11] = 4.0f * C32;                                // C32*y*(4zz-xx-yy)
    m.w[10][11] = -C32;
    m.w[11][11] = -C32;
    m.w[16][12] = 2.0f * C33;                                // C33*z*(2zz-3xx-3yy)
    m.w[14][12] = -3.0f * C33;
    m.w[15][12] = -3.0f * C33;
    m.w[17][13] = 4.0f * C34;                                // C34*x*(4zz-xx-yy)
    m.w[18][13] = -C34;
    m.w[19][13] = -C34;
    m.w[14][14] = C35;                                       // C35*z*(xx-yy)
    m.w[15][14] = -C35;
    m.w[18][15] = C36;                                       // C36*x*(xx-3yy)
    m.w[19][15] = -3.0f * C36;
    return m;
}

__constant__ WMat g_W = makeW();

// ---------------------------------------------------------------------------
// WMMA kernel: each wave32 produces one 16-point x 16-coef tile per loop
// iteration via 5 chained V_WMMA_F32_16X16X4_F32 (K = 20 monomials total).
//
// A fragment (16x4 f32): lane l holds point (l&15); VGPR j holds monomial
// K = 4*i + 2*(l>>4) + j. B fragment (4x16): lane l, VGPR j holds
// W[4*i + 2*(l>>4) + j][l&15]. D: VGPR v, lane l = out[v + 8*(l>>4)][l&15].
// ---------------------------------------------------------------------------
__global__ __launch_bounds__(256) void sh16_wmma_kernel(
    const float* __restrict__ in, const int* __restrict__ size_p,
    float* __restrict__ out, int num_tiles)
{
    const int  lane = threadIdx.x & 31;
    const int  wave = threadIdx.x >> 5;
    const int  n    = lane & 15;        // point-in-tile (A) / coef column (D)
    const bool hi   = lane >= 16;
    const float inv = 1.0f / (float)size_p[0];

    // Constant B fragments, hoisted out of the tile loop.
    v2f b[5];
#pragma unroll
    for (int i = 0; i < 5; ++i) {
        const int krow = 4 * i + (hi ? 2 : 0);
        b[i].x = g_W.w[krow + 0][n];
        b[i].y = g_W.w[krow + 1][n];
    }

    const int waves_per_blk = blockDim.x >> 5;
    const int wave_stride   = gridDim.x * waves_per_blk;

    for (int tile = blockIdx.x * waves_per_blk + wave; tile < num_tiles;
         tile += wave_stride) {
        const int p = tile * 16 + n;    // both half-waves read the same 16 pts
        const float x = in[3 * p + 0] * inv;
        const float y = in[3 * p + 1] * inv;
        const float z = in[3 * p + 2] * inv;
        const float xx = x * x, yy = y * y, zz = z * z;
        const float xy = x * y, yz = y * z, xz = x * z;

        // Monomial pairs per WMMA step (lower half: K=4i,4i+1; upper: 4i+2,4i+3)
        v2f a0 = { hi ? y      : 1.0f,   hi ? z      : x      };
        v2f a1 = { hi ? xz     : xy,     hi ? xx     : yz     };
        v2f a2 = { hi ? xx * y : yy,     hi ? yy * y : zz     };
        v2f a3 = { hi ? xx * z : xy * z, hi ? yy * z : y * zz };
        v2f a4 = { hi ? xx * x : z * zz, hi ? yy * x : x * zz };

        v8f c = {0.f, 0.f, 0.f, 0.f, 0.f, 0.f, 0.f, 0.f};
        c = __builtin_amdgcn_wmma_f32_16x16x4_f32(false, a0, false, b[0], (short)0, c, false, false);
        c = __builtin_amdgcn_wmma_f32_16x16x4_f32(false, a1, false, b[1], (short)0, c, false, false);
        c = __builtin_amdgcn_wmma_f32_16x16x4_f32(false, a2, false, b[2], (short)0, c, false, false);
        c = __builtin_amdgcn_wmma_f32_16x16x4_f32(false, a3, false, b[3], (short)0, c, false, false);
        c = __builtin_amdgcn_wmma_f32_16x16x4_f32(false, a4, false, b[4], (short)0, c, false, false);

        // D[M = v + 8*hi][N = n] -> out[(tile*16 + M)*16 + N], NT (no reuse).
        const size_t base = (size_t)tile * 256 + (hi ? 128u : 0u) + (unsigned)n;
#pragma unroll
        for (int v = 0; v < 8; ++v)
            __builtin_nontemporal_store(c[v], out + base + (size_t)v * 16);
    }
}

// ---------------------------------------------------------------------------
// Scalar tail kernel for point counts not divisible by 16 (not hit for B=4M).
// ---------------------------------------------------------------------------
__global__ void sh16_tail_kernel(const float* __restrict__ in,
                                 const int* __restrict__ size_p,
                                 float* __restrict__ out, int start, int npts)
{
    const int p = start + blockIdx.x * blockDim.x + threadIdx.x;
    if (p >= npts) return;
    const float inv = 1.0f / (float)size_p[0];
    const float x = in[3 * p + 0] * inv;
    const float y = in[3 * p + 1] * inv;
    const float z = in[3 * p + 2] * inv;
    const float xx = x * x, yy = y * y, zz = z * z;
    const float xy = x * y, yz = y * z, xz = x * z;
    float* o = out + (size_t)p * 16;
    o[0]  = 0.28209479177387814f;
    o[1]  = -0.4886025119029199f * y;
    o[2]  = 0.4886025119029199f * z;
    o[3]  = -0.4886025119029199f * x;
    o[4]  = 1.0925484305920792f * xy;
    o[5]  = -1.0925484305920792f * yz;
    o[6]  = 0.31539156525252005f * (2.0f * zz - xx - yy);
    o[7]  = -1.0925484305920792f * xz;
    o[8]  = 0.5462742152960396f * (xx - yy);
    o[9]  = -0.5900435899266435f * y * (3.0f * xx - yy);
    o[10] = 2.890611442640554f * xy * z;
    o[11] = -0.4570457994644658f * y * (4.0f * zz - xx - yy);
    o[12] = 0.3731763325901154f * z * (2.0f * zz - 3.0f * xx - 3.0f * yy);
    o[13] = -0.4570457994644658f * x * (4.0f * zz - xx - yy);
    o[14] = 1.445305721320277f * z * (xx - yy);
    o[15] = -0.5900435899266435f * x * (xx - 3.0f * yy);
}

extern "C" void kernel_launch(void* const* d_in, const int* in_sizes, int n_in,
                              void* d_out, int out_size, void* d_ws, size_t ws_size,
                              hipStream_t stream) {
    const float* in  = (const float*)d_in[0];
    const int*   sz  = (const int*)d_in[1];
    float*       out = (float*)d_out;

    const long npts      = (long)in_sizes[0] / 3;     // B points (x,y,z)
    const int  num_tiles = (int)(npts / 16);

    if (num_tiles > 0) {
        const int waves_per_blk = 8;                  // 256 threads = 8 wave32
        int blocks = (num_tiles + waves_per_blk - 1) / waves_per_blk;
        if (blocks > 8192) blocks = 8192;             // grid-stride over tiles
        sh16_wmma_kernel<<<blocks, 256, 0, stream>>>(in, sz, out, num_tiles);
    }

    const int tail_start = num_tiles * 16;
    const int tail = (int)(npts - tail_start);
    if (tail > 0) {
        sh16_tail_kernel<<<(tail + 255) / 256, 256, 0, stream>>>(
            in, sz, out, tail_start, (int)npts);
    }
}